// GNN_17523466567901
// MI455X (gfx1250) — compile-verified
//
#include <hip/hip_runtime.h>

#define GN 100000            // nodes
#define GE 1600000           // raw edges
#define GET (GE + GN)        // edges + self loops
#define GH 4                 // heads
#define GC 16                // channels per head
#define GHC 64               // H*C

typedef float v2f __attribute__((ext_vector_type(2)));
typedef float v8f __attribute__((ext_vector_type(8)));

// ---- monotone float <-> sortable uint for atomic max over f32 ----
__device__ __forceinline__ unsigned f2key(float v) {
    unsigned b = __float_as_uint(v);
    return (b & 0x80000000u) ? ~b : (b | 0x80000000u);
}
__device__ __forceinline__ float key2f(unsigned k) {
    unsigned b = (k & 0x80000000u) ? (k & 0x7FFFFFFFu) : ~k;
    return __uint_as_float(b);
}

__device__ __forceinline__ void edge_nodes(const int* __restrict__ ei, int e, int& sN, int& dN) {
    if (e < GE) { sN = ei[e]; dN = ei[GE + e]; }
    else        { sN = dN = e - GE; }
}

// ---------------- layer-0 transform: h = x (Nx1) @ W0 (1x64) ----------------
__global__ void k_xform0(const float* __restrict__ x, const float* __restrict__ W0,
                         float* __restrict__ O) {
    int i = blockIdx.x * blockDim.x + threadIdx.x;
    if (i >= GN * GHC) return;
    O[i] = x[i >> 6] * W0[i & 63];
}

// ---------------- f32 WMMA GEMM: [N,64] @ [64,64] -> [N,64] ----------------
// One wave per 16x16 output tile; 4 waves/block cover the 64 output columns.
// A (16x4 f32): lanes 0-15 hold row M=lane, VGPR0/1 = K,K+1; lanes 16-31 hold K+2,K+3.
// B (4x16 f32): lanes 0-15 hold col N=lane, VGPR0/1 = K,K+1; lanes 16-31 hold K+2,K+3.
// C/D (16x16 f32): VGPR r -> M=r (lanes 0-15) / M=r+8 (lanes 16-31), N = lane%16.
__global__ void k_gemm64(const float* __restrict__ A, const float* __restrict__ W,
                         float* __restrict__ O) {
    const int tile = blockIdx.x;          // N/16 = 6250 row tiles
    const int wave = threadIdx.x >> 5;    // 0..3 -> column tile
    const int lane = threadIdx.x & 31;
    const int lh   = lane & 15;
    const int hi   = lane >> 4;           // upper half-wave handles K+2,K+3
    const int row  = tile * 16 + lh;
    const int col0 = wave * 16;

    const float* arow = A + (size_t)row * GHC + hi * 2;
    const float* bcol = W + (size_t)(hi * 2) * GHC + col0 + lh;

    v8f acc = {};
#pragma unroll
    for (int k = 0; k < GHC; k += 4) {
        v2f av, bv;
        av.x = arow[k];
        av.y = arow[k + 1];
        bv.x = bcol[(size_t)k * GHC];
        bv.y = bcol[(size_t)(k + 1) * GHC];
        acc = __builtin_amdgcn_wmma_f32_16x16x4_f32(
            /*neg_a=*/false, av, /*neg_b=*/false, bv,
            /*c_mod=*/(short)0, acc, /*reuse_a=*/false, /*reuse_b=*/false);
    }

    const int ccol = col0 + lh;
    const int crow = tile * 16 + hi * 8;
#pragma unroll
    for (int r = 0; r < 8; ++r)
        O[(size_t)(crow + r) * GHC + ccol] = acc[r];
}

// ---- per-node: es/ed = <h[n,h,:], a_src/a_dst[h,:]>, zero max/sum/acc ----
__global__ void k_nodeinit(const float* __restrict__ Hb, const float* __restrict__ asrc,
                           const float* __restrict__ adst, float* __restrict__ es,
                           float* __restrict__ ed, unsigned* __restrict__ ms,
                           float* __restrict__ ss, float* __restrict__ acc) {
    int t = blockIdx.x * blockDim.x + threadIdx.x;
    if (t >= GN * GH) return;
    int n = t >> 2, hh = t & 3;
    const float4* hv = (const float4*)(Hb + (size_t)n * GHC + hh * GC);
    const float4* av = (const float4*)(asrc + hh * GC);
    const float4* dv = (const float4*)(adst + hh * GC);
    float s = 0.f, d = 0.f;
#pragma unroll
    for (int q = 0; q < 4; ++q) {
        float4 h4 = hv[q], a4 = av[q], d4 = dv[q];
        s += h4.x * a4.x + h4.y * a4.y + h4.z * a4.z + h4.w * a4.w;
        d += h4.x * d4.x + h4.y * d4.y + h4.z * d4.z + h4.w * d4.w;
    }
    es[t] = s; ed[t] = d;
    ms[t] = 0u;            // sortable key <= every real float's key
    ss[t] = 0.f;
    float4* ar = (float4*)(acc + (size_t)n * GHC + hh * GC);
#pragma unroll
    for (int q = 0; q < 4; ++q) ar[q] = make_float4(0.f, 0.f, 0.f, 0.f);
}

__device__ __forceinline__ float edge_e(const float* es, const float* ed,
                                        int sN, int dN, int hh) {
    float v = es[sN * GH + hh] + ed[dN * GH + hh];
    return v >= 0.f ? v : 0.2f * v;       // leaky relu
}

__global__ void k_edge_max(const int* __restrict__ ei, const float* __restrict__ es,
                           const float* __restrict__ ed, unsigned* __restrict__ ms) {
    int t = blockIdx.x * blockDim.x + threadIdx.x;
    if (t >= GET * GH) return;
    int e = t >> 2, hh = t & 3;
    int sN, dN; edge_nodes(ei, e, sN, dN);
    atomicMax(&ms[dN * GH + hh], f2key(edge_e(es, ed, sN, dN, hh)));
}

__global__ void k_edge_sum(const int* __restrict__ ei, const float* __restrict__ es,
                           const float* __restrict__ ed, const unsigned* __restrict__ ms,
                           float* __restrict__ ss) {
    int t = blockIdx.x * blockDim.x + threadIdx.x;
    if (t >= GET * GH) return;
    int e = t >> 2, hh = t & 3;
    int sN, dN; edge_nodes(ei, e, sN, dN);
    float v = edge_e(es, ed, sN, dN, hh);
    float m = key2f(ms[dN * GH + hh]);
    atomicAdd(&ss[dN * GH + hh], __expf(v - m));
}

__global__ void k_edge_agg(const int* __restrict__ ei, const float* __restrict__ es,
                           const float* __restrict__ ed, const unsigned* __restrict__ ms,
                           const float* __restrict__ ss, const float* __restrict__ Hb,
                           float* __restrict__ acc) {
    int t = blockIdx.x * blockDim.x + threadIdx.x;
    if (t >= GET * GH) return;
    int e = t >> 2, hh = t & 3;
    int sN, dN; edge_nodes(ei, e, sN, dN);
    float v = edge_e(es, ed, sN, dN, hh);
    float m = key2f(ms[dN * GH + hh]);
    float alpha = __expf(v - m) / (ss[dN * GH + hh] + 1e-16f);
    const float4* hv = (const float4*)(Hb + (size_t)sN * GHC + hh * GC);
    float* ar = acc + (size_t)dN * GHC + hh * GC;
#pragma unroll
    for (int q = 0; q < 4; ++q) {
        float4 h4 = hv[q];
        atomicAdd(&ar[q * 4 + 0], alpha * h4.x);
        atomicAdd(&ar[q * 4 + 1], alpha * h4.y);
        atomicAdd(&ar[q * 4 + 2], alpha * h4.z);
        atomicAdd(&ar[q * 4 + 3], alpha * h4.w);
    }
}

__global__ void k_bias(float* __restrict__ acc, const float* __restrict__ bias) {
    int i = blockIdx.x * blockDim.x + threadIdx.x;
    if (i >= GN * GHC) return;
    acc[i] += bias[i & 63];
}

// ---------------- latent MLP ----------------
__global__ void k_mlp1(const float* __restrict__ in, const float* __restrict__ lw1,
                       const float* __restrict__ lb1, float* __restrict__ tb) {
    int t = blockIdx.x * blockDim.x + threadIdx.x;
    if (t >= GN * GHC) return;
    int n = t >> 6, j = t & 63;
    const float* r = in + (size_t)n * GHC;
    float a = lb1[j];
#pragma unroll 8
    for (int k = 0; k < GHC; ++k) a += r[k] * lw1[k * GHC + j];
    tb[t] = a > 0.f ? a : 0.f;
}

__global__ void k_mlp2(const float* __restrict__ tb, const float* __restrict__ lw2,
                       const float* __restrict__ lb2, const float* __restrict__ dw,
                       const float* __restrict__ db, float* __restrict__ out) {
    int n = blockIdx.x * blockDim.x + threadIdx.x;
    if (n >= GN) return;
    float tr[GHC];
    const float4* tv = (const float4*)(tb + (size_t)n * GHC);
#pragma unroll
    for (int q = 0; q < 16; ++q) {
        float4 f = tv[q];
        tr[4 * q + 0] = f.x; tr[4 * q + 1] = f.y;
        tr[4 * q + 2] = f.z; tr[4 * q + 3] = f.w;
    }
    float u[16];
#pragma unroll
    for (int j = 0; j < 16; ++j) {
        float a = lb2[j];
#pragma unroll
        for (int k = 0; k < GHC; ++k) a += tr[k] * lw2[k * 16 + j];
        u[j] = a;
    }
    float lg[4];
#pragma unroll
    for (int j = 0; j < 4; ++j) {
        float a = db[j];
#pragma unroll
        for (int k = 0; k < 16; ++k) a += u[k] * dw[k * 4 + j];
        lg[j] = a;
    }
    float mx = fmaxf(fmaxf(lg[0], lg[1]), fmaxf(lg[2], lg[3]));
    float s = 0.f;
#pragma unroll
    for (int j = 0; j < 4; ++j) { lg[j] = __expf(lg[j] - mx); s += lg[j]; }
    float inv = 1.f / s;
#pragma unroll
    for (int j = 0; j < 4; ++j) out[(size_t)n * 4 + j] = lg[j] * inv;
}

extern "C" void kernel_launch(void* const* d_in, const int* in_sizes, int n_in,
                              void* d_out, int out_size, void* d_ws, size_t ws_size,
                              hipStream_t stream) {
    (void)in_sizes; (void)n_in; (void)out_size; (void)ws_size;
    const float* x   = (const float*)d_in[0];
    const int*   ei  = (const int*)d_in[1];
    const float* W[3]  = {(const float*)d_in[2],  (const float*)d_in[6],  (const float*)d_in[10]};
    const float* AS[3] = {(const float*)d_in[3],  (const float*)d_in[7],  (const float*)d_in[11]};
    const float* AD[3] = {(const float*)d_in[4],  (const float*)d_in[8],  (const float*)d_in[12]};
    const float* B[3]  = {(const float*)d_in[5],  (const float*)d_in[9],  (const float*)d_in[13]};
    const float* lw1 = (const float*)d_in[14];
    const float* lb1 = (const float*)d_in[15];
    const float* lw2 = (const float*)d_in[16];
    const float* lb2 = (const float*)d_in[17];
    const float* dw  = (const float*)d_in[18];
    const float* db  = (const float*)d_in[19];
    float* out = (float*)d_out;

    // workspace layout (fits ~83 MB; whole working set ~L2 sized)
    size_t nhc = (size_t)GN * GHC;
    float*    bufA = (float*)d_ws;         // ping (layer outputs 0, 2)
    float*    bufH = bufA + nhc;           // transformed features h, reused as MLP temp
    float*    bufC = bufH + nhc;           // pong (layer 1 output)
    float*    es   = bufC + nhc;
    float*    ed   = es + (size_t)GN * GH;
    float*    ss   = ed + (size_t)GN * GH;
    unsigned* ms   = (unsigned*)(ss + (size_t)GN * GH);

    const int TB = 256;
    const int gridNHC = (GN * GHC + TB - 1) / TB;
    const int gridNH  = (GN * GH  + TB - 1) / TB;
    const int gridEH  = (GET * GH + TB - 1) / TB;

    const float* in = nullptr;
    float* accs[3] = {bufA, bufC, bufA};
    for (int l = 0; l < 3; ++l) {
        if (l == 0) k_xform0<<<gridNHC, TB, 0, stream>>>(x, W[0], bufH);
        else        k_gemm64<<<GN / 16, 128, 0, stream>>>(in, W[l], bufH);
        float* acc = accs[l];
        k_nodeinit<<<gridNH, TB, 0, stream>>>(bufH, AS[l], AD[l], es, ed, ms, ss, acc);
        k_edge_max<<<gridEH, TB, 0, stream>>>(ei, es, ed, ms);
        k_edge_sum<<<gridEH, TB, 0, stream>>>(ei, es, ed, ms, ss);
        k_edge_agg<<<gridEH, TB, 0, stream>>>(ei, es, ed, ms, ss, bufH, acc);
        k_bias<<<gridNHC, TB, 0, stream>>>(acc, B[l]);
        in = acc;
    }
    k_mlp1<<<gridNHC, TB, 0, stream>>>(in, lw1, lb1, bufH);
    k_mlp2<<<(GN + TB - 1) / TB, TB, 0, stream>>>(bufH, lw2, lb2, dw, db, out);
}